// GCN_Module_41437844472039
// MI455X (gfx1250) — compile-verified
//
#include <hip/hip_runtime.h>
#include <hip/hip_bf16.h>

typedef __attribute__((ext_vector_type(2))) float v2f;
typedef __attribute__((ext_vector_type(4))) float v4f;
typedef __attribute__((ext_vector_type(8))) float v8f;

#define NBOX 512
#define NFG  1024
#define NM1  511   // NBOX - 1

// ---------------------------------------------------------------------------
// Kernel 1: agg[p,f] = adj[p,0]*gbf[p,f] + sum_j adj[p,1+j]*rpf[p,j,f]
// Pure HBM-bandwidth streaming (~1.07 GB of rpf, read exactly once).
// One block per node p, 256 threads x float4 = one full 1024-feature row per
// iteration. adj[p,1+j] is block-uniform -> s_load broadcast.
// rpf loads are NON-TEMPORAL (TH=NT): keep the read-once stream from evicting
// W_gcn / agg out of the 192 MB L2 before kernel 2 consumes them.
// ---------------------------------------------------------------------------
__global__ __launch_bounds__(256)
void agg_kernel(const float* __restrict__ gbf,
                const float* __restrict__ adj,
                const float* __restrict__ rpf,
                float* __restrict__ agg)
{
    const int p  = blockIdx.x;
    const int f4 = threadIdx.x;                 // float4 index 0..255

    const v4f* __restrict__ g4 = (const v4f*)(gbf + (size_t)p * NFG);
    const v4f* __restrict__ r4 = (const v4f*)(rpf + (size_t)p * NM1 * NFG);
    const float* __restrict__ arow = adj + (size_t)p * NBOX;

    v4f acc = g4[f4] * arow[0];

    for (int j = 0; j < NM1; ++j) {
        // prefetch a few rows ahead (speculative; OOB at tail is dropped)
        __builtin_prefetch(&r4[(size_t)(j + 12) * (NFG / 4) + f4], 0, 0);
        const float w = arow[1 + j];            // uniform -> scalar load
        const v4f  r = __builtin_nontemporal_load(&r4[(size_t)j * (NFG / 4) + f4]);
        acc += r * w;
    }

    // agg is re-read by kernel 2 almost immediately -> regular (L2-resident) store.
    ((v4f*)(agg + (size_t)p * NFG))[f4] = acc;
}

// ---------------------------------------------------------------------------
// Kernel 2 (fused): x = agg @ W^T, then LayerNorm(1024) + ReLU, per 16-row
// strip. 8 waves; each wave computes 8 WMMA 16x16 tiles (128 cols) with
// V_WMMA_F32_16X16X4_F32, K-loop step 4. Result tile staged in 64KB LDS,
// then wave32 shuffle reductions do mean/var (2 rows per wave).
//
// A fragment (16x4 f32): lanes 0-15 row M=lane, v0=K0,v1=K1;
//                        lanes 16-31 row M=lane-16, v0=K2,v1=K3.
// B fragment (4x16 f32): v0 = {K0 | K2} x N=0..15, v1 = {K1 | K3}.
// C/D (16x16 f32): VGPR v -> row v (lanes 0-15) / row v+8 (lanes 16-31).
// ---------------------------------------------------------------------------
__global__ __launch_bounds__(256)
void gemm_ln_kernel(const float* __restrict__ A,     // agg [512,1024]
                    const float* __restrict__ W,     // W_gcn [1024,1024]
                    const float* __restrict__ gamma,
                    const float* __restrict__ beta,
                    float* __restrict__ out)         // [512,1024]
{
    __shared__ float xbuf[16 * NFG];                 // 64 KB strip of x

    const int r0      = blockIdx.x * 16;
    const int lane    = threadIdx.x & 31;
    const int wave    = threadIdx.x >> 5;
    const int half    = lane >> 4;                   // 0: K0/K1, 1: K2/K3
    const int l16     = lane & 15;
    const int colBase = wave * 128;                  // 8 tiles of 16 cols

    v8f acc[8] = {};                                 // zero-init accumulators

    const float* __restrict__ Arow = A + (size_t)(r0 + l16) * NFG;

    for (int k = 0; k < NFG; k += 4) {
        const int koff = k + half * 2;
        const v2f a = *(const v2f*)(Arow + koff);    // A[r0+l16][koff..koff+1]
#pragma unroll
        for (int t = 0; t < 8; ++t) {
            const int n = colBase + t * 16 + l16;    // output column
            const v2f b = *(const v2f*)(W + (size_t)n * NFG + koff); // B[k][n]=W[n][k]
            acc[t] = __builtin_amdgcn_wmma_f32_16x16x4_f32(
                false, a, false, b, (short)0, acc[t], false, false);
        }
    }

    // Spill the 16x1024 strip of x into LDS for the row-wise LayerNorm.
#pragma unroll
    for (int t = 0; t < 8; ++t) {
        const int col = colBase + t * 16 + l16;
#pragma unroll
        for (int v = 0; v < 8; ++v) {
            const int row = v + half * 8;
            xbuf[row * NFG + col] = acc[t][v];
        }
    }
    __syncthreads();

    // LayerNorm + ReLU: wave handles rows 2*wave and 2*wave+1.
    for (int rr = 0; rr < 2; ++rr) {
        const int r = wave * 2 + rr;
        float s = 0.f, ss = 0.f;
#pragma unroll
        for (int i = 0; i < 32; ++i) {
            const float v = xbuf[r * NFG + lane + 32 * i];
            s  += v;
            ss += v * v;
        }
#pragma unroll
        for (int off = 16; off > 0; off >>= 1) {     // wave32 butterfly
            s  += __shfl_xor(s,  off, 32);
            ss += __shfl_xor(ss, off, 32);
        }
        const float mu   = s  * (1.0f / NFG);
        const float var  = ss * (1.0f / NFG) - mu * mu;
        const float rstd = rsqrtf(var + 1e-5f);

        float* __restrict__ orow = out + (size_t)(r0 + r) * NFG;
#pragma unroll
        for (int i = 0; i < 32; ++i) {
            const int c = lane + 32 * i;
            float v = (xbuf[r * NFG + c] - mu) * rstd * gamma[c] + beta[c];
            // final result is never re-read on device -> non-temporal store
            __builtin_nontemporal_store(fmaxf(v, 0.0f), &orow[c]);
        }
    }
}

// ---------------------------------------------------------------------------
extern "C" void kernel_launch(void* const* d_in, const int* in_sizes, int n_in,
                              void* d_out, int out_size, void* d_ws, size_t ws_size,
                              hipStream_t stream)
{
    const float* gbf   = (const float*)d_in[0];   // [1,512,1024]
    // d_in[1] = boxes_in_flat : dead path, unused
    const float* adj   = (const float*)d_in[2];   // [512,512]
    const float* rpf   = (const float*)d_in[3];   // [512*511,1024]
    const float* W     = (const float*)d_in[4];   // [1024,1024]
    const float* gamma = (const float*)d_in[5];   // [1024]
    const float* beta  = (const float*)d_in[6];   // [1024]
    float* out = (float*)d_out;                   // [1,512,1024]

    const size_t aggBytes = (size_t)NBOX * NFG * sizeof(float);
    // Prefer scratch; aliasing d_out is safe (each GEMM workgroup reads only
    // the agg rows it later overwrites, strictly before writing them).
    float* agg = (ws_size >= aggBytes) ? (float*)d_ws : out;

    agg_kernel<<<NBOX, 256, 0, stream>>>(gbf, adj, rpf, agg);
    gemm_ln_kernel<<<NBOX / 16, 256, 0, stream>>>(agg, W, gamma, beta, out);
}